// LI_no_Spike_17806934409275
// MI455X (gfx1250) — compile-verified
//
#include <hip/hip_runtime.h>

// LI (leaky integrator, no spike) scan:  u_t = clamp(tau,0,1)*u_{t-1} + x_t
// x:[B=64,F=4096,T=128] f32, tau:[F] f32, out:[B,F,T] f32.
// Memory-bound: 256 MiB @ 23.3 TB/s ~ 11.5 us. Compute done as chunked
// triangular matmuls on the f32 WMMA path (exact fp32 math).

typedef __attribute__((ext_vector_type(2))) float v2f;
typedef __attribute__((ext_vector_type(8))) float v8f;

#if defined(__HIP_DEVICE_COMPILE__) && defined(__gfx1250__)
#if !__has_builtin(__builtin_amdgcn_wmma_f32_16x16x4_f32)
#error "missing __builtin_amdgcn_wmma_f32_16x16x4_f32 on gfx1250"
#endif
#endif

namespace {
constexpr int B = 64;
constexpr int F = 4096;
constexpr int T = 128;
constexpr int LDS_STRIDE = 132;  // floats; multiple of 4 -> 16B-aligned rows, bank-spread
}

__global__ __launch_bounds__(128) void li_scan_wmma(const float* __restrict__ x,
                                                    const float* __restrict__ tau,
                                                    float* __restrict__ out) {
    __shared__ float xs[B * LDS_STRIDE];  // 64 x 132 f32 tile (33,792 B), reused for output
    __shared__ float pow_tau[17];         // tau^0 .. tau^16
    __shared__ float carry[B];            // u at end of previous chunk, per row b

    const int f    = blockIdx.x;
    const int tid  = threadIdx.x;
    const int wave = tid >> 5;    // 4 waves
    const int lane = tid & 31;
    const int half = lane >> 4;   // 0: lanes 0-15, 1: lanes 16-31
    const int n    = lane & 15;

    // ---- tau powers (clamped to [0,1] per 'forward' constraint) ----
    if (tid == 0) {
        float tc = tau[f];
        tc = fminf(fmaxf(tc, 0.0f), 1.0f);
        float p = 1.0f;
        #pragma unroll
        for (int i = 0; i <= 16; ++i) { pow_tau[i] = p; p *= tc; }
    }
    if (tid < B) carry[tid] = 0.0f;

    // ---- stage x tile: wave w loads whole contiguous 512B rows (coalesced) ----
    #pragma unroll
    for (int it = 0; it < 16; ++it) {
        const int row = it * 4 + wave;
        const float4* src = reinterpret_cast<const float4*>(x + ((size_t)row * F + f) * (size_t)T);
        float4 v = src[lane];
        *reinterpret_cast<float4*>(&xs[row * LDS_STRIDE + lane * 4]) = v;
    }
    __syncthreads();

    // ---- per-wave B-tile scan: out(16x128) = chunked X(16x16) * M16 + carry path ----
    const int rowBase = wave * 16;  // this wave's 16 rows of b

    // B fragments of M16[s][t] = (s<=t) ? tau^(t-s) : 0, K-step layout: K = 2*half + j
    v2f bfrag[4];
    #pragma unroll
    for (int kk = 0; kk < 4; ++kk) {
        const int s0 = kk * 4 + 2 * half;
        float b0 = (s0     <= n) ? pow_tau[n - s0]     : 0.0f;
        float b1 = (s0 + 1 <= n) ? pow_tau[n - s0 - 1] : 0.0f;
        bfrag[kk] = v2f{b0, b1};
    }
    // carry K-step: single K row with coefficient tau^(t_local+1)
    const v2f bcarry = v2f{(half == 0) ? pow_tau[n + 1] : 0.0f, 0.0f};

    for (int c = 0; c < 8; ++c) {
        const int colBase = c * 16;

        // carry contribution: A row M=n holds carry[b] at K=0 only
        float cv = carry[rowBase + n];           // full-EXEC broadcast load
        v2f acar = v2f{(half == 0) ? cv : 0.0f, 0.0f};

        v8f acc = {};
        acc = __builtin_amdgcn_wmma_f32_16x16x4_f32(false, acar, false, bcarry,
                                                    (short)0, acc, false, false);
        #pragma unroll
        for (int kk = 0; kk < 4; ++kk) {
            const int s = colBase + kk * 4 + 2 * half;     // A layout: K = 2*half + j
            v2f a = *reinterpret_cast<const v2f*>(&xs[(rowBase + n) * LDS_STRIDE + s]);
            acc = __builtin_amdgcn_wmma_f32_16x16x4_f32(false, a, false, bfrag[kk],
                                                        (short)0, acc, false, false);
        }

        // new carry = u at t_local = 15 (column N=15 lives in lanes 15 and 31)
        if (n == 15) {
            #pragma unroll
            for (int r = 0; r < 8; ++r)
                carry[rowBase + 8 * half + r] = acc[r];
        }

        // write output tile into the just-consumed LDS columns (same wave, same rows)
        #pragma unroll
        for (int r = 0; r < 8; ++r)
            xs[(rowBase + r + 8 * half) * LDS_STRIDE + colBase + n] = acc[r];
    }

    __syncthreads();

    // ---- coalesced store of the 64x128 output tile ----
    #pragma unroll
    for (int it = 0; it < 16; ++it) {
        const int row = it * 4 + wave;
        float4 v = *reinterpret_cast<const float4*>(&xs[row * LDS_STRIDE + lane * 4]);
        reinterpret_cast<float4*>(out + ((size_t)row * F + f) * (size_t)T)[lane] = v;
    }
}

extern "C" void kernel_launch(void* const* d_in, const int* in_sizes, int n_in,
                              void* d_out, int out_size, void* d_ws, size_t ws_size,
                              hipStream_t stream) {
    (void)in_sizes; (void)n_in; (void)d_ws; (void)ws_size; (void)out_size;
    const float* x   = reinterpret_cast<const float*>(d_in[0]);   // [B,F,T] f32
    const float* tau = reinterpret_cast<const float*>(d_in[1]);   // [F]     f32
    float* out = reinterpret_cast<float*>(d_out);                 // [B,F,T] f32

    li_scan_wmma<<<dim3(F), dim3(128), 0, stream>>>(x, tau, out);
}